// Quantization_29729763622995
// MI455X (gfx1250) — compile-verified
//
#include <hip/hip_runtime.h>
#include <hip/hip_bf16.h>

typedef __attribute__((ext_vector_type(2))) float v2f;
typedef __attribute__((ext_vector_type(8))) float v8f;

#define DIMS   256
#define KCODES 1024
#define NVEC   32768          // 32 * 32 * 32 spatial vectors
#define BSTRIDE 262144        // 256 * 1024 floats per batch in x
#define BT_STRIDE 260         // padded LDS row stride (floats): 16B-aligned rows, conflict-free b64 reads

// ---------------------------------------------------------------------------
// Kernel 0: wsq[k] = sum_j w[k][j]^2   (one wave per code)
// ---------------------------------------------------------------------------
__global__ __launch_bounds__(256) void vq_wsq_kernel(const float* __restrict__ w,
                                                     float* __restrict__ wsq) {
    const int wave = threadIdx.x >> 5;
    const int lane = threadIdx.x & 31;
    const int code = blockIdx.x * 8 + wave;
    const float* row = w + code * DIMS;
    float s = 0.0f;
#pragma unroll
    for (int i = 0; i < 8; ++i) {
        float v = row[lane + 32 * i];   // coalesced 128B per step
        s += v * v;
    }
#pragma unroll
    for (int off = 16; off > 0; off >>= 1)
        s += __shfl_xor(s, off, 32);
    if (lane == 0) wsq[code] = s;
}

// ---------------------------------------------------------------------------
// Kernel 1: argmin_k ( wsq[k] - 2 * e_n . w_k )  via V_WMMA_F32_16X16X4_F32
// One wave per 16-row strip; A operand held in 128 VGPRs; B tiles staged in
// LDS and shared by the 8 waves of the block.
// ---------------------------------------------------------------------------
__global__ __launch_bounds__(256) void vq_argmin_kernel(const float* __restrict__ x,
                                                        const float* __restrict__ w,
                                                        const float* __restrict__ wsq,
                                                        float* __restrict__ out_idx_f,
                                                        int* __restrict__ ws_idx) {
    __shared__ float bt[16 * BT_STRIDE];   // B tile: [code_in_tile][dim], padded

    const int wave = threadIdx.x >> 5;
    const int lane = threadIdx.x & 31;
    const int m    = lane & 15;            // row (A) / column (B) within tile
    const int half = lane >> 4;

    const int strip  = blockIdx.x * 8 + wave;   // 2048 strips of 16 rows
    const int n_base = strip * 16;
    const int b      = n_base >> 10;            // batch (strips never cross batches)
    const int sp     = (n_base & 1023) + m;     // spatial position for this lane's row
    const float* xb  = x + (size_t)b * BSTRIDE + sp;

    // Preload A fragments: a[2kk],a[2kk+1] = E[m][4kk+2*half +0/+1]
    float a[128];
#pragma unroll
    for (int kk = 0; kk < 64; ++kk) {
        int k = 4 * kk + 2 * half;
        a[2 * kk]     = xb[(size_t)k * 1024];
        a[2 * kk + 1] = xb[(size_t)(k + 1) * 1024];
    }

    float bestv[8];
    int   besti[8];
#pragma unroll
    for (int j = 0; j < 8; ++j) { bestv[j] = 3.402823466e+38f; besti[j] = 0; }

    const int code = threadIdx.x >> 4;     // cooperative B-tile fill: 16 codes x 16 segs
    const int seg  = threadIdx.x & 15;

    for (int t = 0; t < 64; ++t) {         // 64 column tiles of 16 codes
        __syncthreads();                   // previous tile fully consumed
        {
            const float4* src = (const float4*)(w + (size_t)(t * 16 + code) * DIMS + seg * 16);
            float4 v0 = src[0], v1 = src[1], v2 = src[2], v3 = src[3];
            float4* dst = (float4*)&bt[code * BT_STRIDE + seg * 16];   // 16B-aligned (260*4=1040)
            dst[0] = v0; dst[1] = v1; dst[2] = v2; dst[3] = v3;
        }
        __syncthreads();

        v8f acc = {0.f, 0.f, 0.f, 0.f, 0.f, 0.f, 0.f, 0.f};
#pragma unroll
        for (int kk = 0; kk < 64; ++kk) {
            int k = 4 * kk + 2 * half;
            v2f av = {a[2 * kk], a[2 * kk + 1]};
            v2f bv = *(const v2f*)&bt[m * BT_STRIDE + k];   // 8B-aligned, conflict-free
            acc = __builtin_amdgcn_wmma_f32_16x16x4_f32(
                false, av, false, bv, (short)0, acc, false, false);
        }

        const int   n  = t * 16 + m;            // column (code) this lane holds
        const float wq = wsq[n];
#pragma unroll
        for (int j = 0; j < 8; ++j) {           // VGPR j <-> row M = j + 8*half
            float s = wq - 2.0f * acc[j];
            if (s < bestv[j]) { bestv[j] = s; besti[j] = n; }  // strict < => first index wins over t
        }
    }

    // Reduce (val,idx) across the 16 lanes of each half-wave; first-index tie-break.
#pragma unroll
    for (int j = 0; j < 8; ++j) {
        float v   = bestv[j];
        int   idx = besti[j];
#pragma unroll
        for (int off = 1; off < 16; off <<= 1) {
            float ov = __shfl_xor(v, off, 32);
            int   oi = __shfl_xor(idx, off, 32);
            if (ov < v || (ov == v && oi < idx)) { v = ov; idx = oi; }
        }
        if (m == 0) {
            int n_row = n_base + j + 8 * half;
            ws_idx[n_row]    = idx;
            out_idx_f[n_row] = (float)idx;
        }
    }
}

// ---------------------------------------------------------------------------
// Kernel 2: produce encoded_flat, quantized_flat, quantized (NCHW) with all
// global accesses coalesced via two padded LDS transpose tiles.
// Block = 256 threads handles a 32(c) x 32(spatial) tile of one batch.
// ---------------------------------------------------------------------------
__global__ __launch_bounds__(256) void vq_outputs_kernel(const float* __restrict__ x,
                                                         const float* __restrict__ w,
                                                         const int* __restrict__ ws_idx,
                                                         float* __restrict__ out_enc,
                                                         float* __restrict__ out_qf,
                                                         float* __restrict__ out_q) {
    __shared__ float te[32][33];   // encoded tile, [c_local][sp_local]
    __shared__ float tq[32][33];   // quantized tile, [c_local][sp_local]

    const int tx  = threadIdx.x & 31;
    const int ty  = threadIdx.x >> 5;
    const int bid = blockIdx.x;
    const int b   = bid >> 8;
    const int rem = bid & 255;
    const int c0  = (rem >> 5) * 32;   // 8 channel tiles
    const int sp0 = (rem & 31) * 32;   // 32 spatial tiles

    // Phase A: read x coalesced along spatial
#pragma unroll
    for (int yy = 0; yy < 4; ++yy) {
        int cl = ty + 8 * yy;
        te[cl][tx] = x[(size_t)b * BSTRIDE + (size_t)(c0 + cl) * 1024 + sp0 + tx];
    }
    __syncthreads();

    // Phase B: n-major, coalesced along channel for the [N,D] outputs
#pragma unroll
    for (int yy = 0; yy < 4; ++yy) {
        int nl = ty + 8 * yy;
        int n  = b * 1024 + sp0 + nl;
        int c  = c0 + tx;
        float e = te[tx][nl];                       // stride-33: conflict-free
        out_enc[(size_t)n * DIMS + c] = e;
        int idx = ws_idx[n];                        // scalar per wave -> coalesced gather
        float q = w[(size_t)idx * DIMS + c];
        out_qf[(size_t)n * DIMS + c] = q;
        tq[tx][nl] = q;
    }
    __syncthreads();

    // Phase C: write quantized back in NCHW, coalesced along spatial
#pragma unroll
    for (int yy = 0; yy < 4; ++yy) {
        int cl = ty + 8 * yy;
        out_q[(size_t)b * BSTRIDE + (size_t)(c0 + cl) * 1024 + sp0 + tx] = tq[cl][tx];
    }
}

// ---------------------------------------------------------------------------
extern "C" void kernel_launch(void* const* d_in, const int* in_sizes, int n_in,
                              void* d_out, int out_size, void* d_ws, size_t ws_size,
                              hipStream_t stream) {
    const float* x = (const float*)d_in[0];   // [32,256,32,32]
    const float* w = (const float*)d_in[1];   // [1024,256]

    float* out_enc  = (float*)d_out;                       // [32768,256]
    float* out_qf   = out_enc + (size_t)NVEC * DIMS;       // [32768,256]
    float* out_idxf = out_enc + 2 * (size_t)NVEC * DIMS;   // [32768]
    float* out_q    = out_idxf + NVEC;                     // [32,256,32,32]

    float* wsq   = (float*)d_ws;                           // 1024 floats
    int*   wsidx = (int*)((char*)d_ws + KCODES * sizeof(float));  // 32768 ints

    vq_wsq_kernel<<<KCODES / 8, 256, 0, stream>>>(w, wsq);
    vq_argmin_kernel<<<NVEC / 16 / 8, 256, 0, stream>>>(x, w, wsq, out_idxf, wsidx);
    vq_outputs_kernel<<<32 * 8 * 32, 256, 0, stream>>>(x, w, wsidx, out_enc, out_qf, out_q);
}